// ODEFunc_70385924047276
// MI455X (gfx1250) — compile-verified
//
#include <hip/hip_runtime.h>

#define HID 128            // hidden width (floats per node row)
#define CHUNKS_PER_NODE 32 // HID/4 float4 chunks
#define ROW_BYTES 512      // HID * 4 bytes

typedef float float4v __attribute__((ext_vector_type(4)));

// ---------------------------------------------------------------------------
// Kernel 1: zero the accumulator (d_out doubles as ax). float4 stores.
// ---------------------------------------------------------------------------
__global__ void odefunc_zero(float4v* __restrict__ p, int n4) {
    int i = blockIdx.x * blockDim.x + threadIdx.x;
    int stride = gridDim.x * blockDim.x;
    for (; i < n4; i += stride) {
        float4v z = {0.f, 0.f, 0.f, 0.f};
        p[i] = z;
    }
}

// ---------------------------------------------------------------------------
// Async gather helper: per-lane 16B global -> LDS DMA (gfx1250, ASYNCcnt).
// lds_addr is the low-32-bit LDS byte offset; gaddr a 64-bit global address.
// ---------------------------------------------------------------------------
__device__ __forceinline__ void async_gather_b128(unsigned int lds_addr,
                                                  unsigned long long gaddr) {
    asm volatile("global_load_async_to_lds_b128 %0, %1, off"
                 :: "v"(lds_addr), "v"(gaddr)
                 : "memory");
}

// ---------------------------------------------------------------------------
// Kernel 2: COO SpMM scatter with an async-DMA double-buffered row gather.
// One wave32 per edge slot: lane l owns columns [4l, 4l+4) of the 128-wide
// row. The x[col] gather (512B, L2-resident on MI455X: x is 51.2MB < 192MB
// L2) is issued as GLOBAL_LOAD_ASYNC_TO_LDS_B128 one iteration ahead, so the
// DMA overlaps the 128 atomic-dword scatter of the current edge. Segment-sum
// is done with non-returning hardware global_atomic_add_f32 (STOREcnt path).
// ---------------------------------------------------------------------------
__global__ void odefunc_spmm_scatter(const float* __restrict__ vals,
                                     const int*   __restrict__ rows,
                                     const int*   __restrict__ cols,
                                     const float* __restrict__ x,
                                     float*       __restrict__ ax,
                                     int n_edges) {
    // 8 waves/block, 2 x 512B buffers per wave
    __shared__ __align__(16) char smem[8 * 2 * ROW_BYTES];

    const int lane = threadIdx.x & 31;
    const int wid  = threadIdx.x >> 5;           // wave id within block
    const int wave = (blockIdx.x * blockDim.x + threadIdx.x) >> 5;
    const int wstride = (gridDim.x * blockDim.x) >> 5;

    // Per-wave LDS double buffer (byte offsets inside LDS space).
    char* buf0 = &smem[(wid * 2 + 0) * ROW_BYTES];
    char* buf1 = &smem[(wid * 2 + 1) * ROW_BYTES];
    const unsigned int lds0 =
        (unsigned int)(unsigned long long)(buf0 + lane * 16);
    const unsigned int lds1 =
        (unsigned int)(unsigned long long)(buf1 + lane * 16);

    int e = wave;
    if (e >= n_edges) return;

    // Prologue: start the DMA for the first edge's row.
    {
        const int c0 = cols[__builtin_amdgcn_readfirstlane(e)];
        async_gather_b128(lds0,
            (unsigned long long)(x + (size_t)c0 * HID + lane * 4));
    }

    int sel = 0;
    for (; e < n_edges; e += wstride) {
        const int eu  = __builtin_amdgcn_readfirstlane(e);
        const int row = rows[eu];
        const float v = vals[eu];

        const int en = e + wstride;
        if (en < n_edges) {
            // Next row's DMA into the other buffer. Ensure the previous
            // ds_load of that buffer has fully drained (WAR vs DMA write).
            const int cn = cols[__builtin_amdgcn_readfirstlane(en)];
            asm volatile("s_wait_dscnt 0x0" ::: "memory");
            async_gather_b128(sel ? lds0 : lds1,
                (unsigned long long)(x + (size_t)cn * HID + lane * 4));
            // Two async ops outstanding; <=1 means the older (current
            // buffer's) DMA has completed — async loads finish in order.
            asm volatile("s_wait_asynccnt 0x1" ::: "memory");
        } else {
            asm volatile("s_wait_asynccnt 0x0" ::: "memory");
        }

        // Read the staged row from LDS and scatter-add.
        const float4v xv =
            *(const float4v*)((sel ? buf1 : buf0) + lane * 16);
        const float4v m = xv * v;

        float* dst = ax + (size_t)row * HID + lane * 4;
        unsafeAtomicAdd(dst + 0, m.x);
        unsafeAtomicAdd(dst + 1, m.y);
        unsafeAtomicAdd(dst + 2, m.z);
        unsafeAtomicAdd(dst + 3, m.w);

        sel ^= 1;
    }
}

// ---------------------------------------------------------------------------
// Kernel 3: epilogue, in place on d_out:
//   out = sigmoid(alpha[n]) * 0.5 * (ax - x) + x0
// One thread per float4 chunk; 32 chunks per node, so alpha[n] is
// wave-uniform — force the scalar path with readfirstlane.
// ---------------------------------------------------------------------------
__global__ void odefunc_finalize(const float* __restrict__ x,
                                 const float* __restrict__ x0,
                                 const float* __restrict__ alpha,
                                 float*       __restrict__ out,
                                 int n_nodes) {
    const int i = blockIdx.x * blockDim.x + threadIdx.x;
    const int total = n_nodes * CHUNKS_PER_NODE;
    if (i >= total) return;

    const int n = __builtin_amdgcn_readfirstlane(i >> 5);
    const float a = alpha[n];
    const float s = 0.5f / (1.0f + __expf(-a)); // 0.5 * sigmoid(a)

    const float4v xv  = ((const float4v*)x)[i];
    const float4v x0v = ((const float4v*)x0)[i];
    const float4v axv = ((float4v*)out)[i];

    const float4v f = (axv - xv) * s + x0v;
    ((float4v*)out)[i] = f;
}

// ---------------------------------------------------------------------------
// Host-side launcher. Input order (from setup_inputs):
//   0: x [N*H] f32, 1: x0 [N*H] f32, 2: alpha_train [N] f32,
//   3: adj_vals [E] f32, 4: adj_rows [E] i32, 5: adj_cols [E] i32
// d_out: [N*H] f32. Used as the ax accumulator, then finalized in place.
// ---------------------------------------------------------------------------
extern "C" void kernel_launch(void* const* d_in, const int* in_sizes, int n_in,
                              void* d_out, int out_size, void* d_ws, size_t ws_size,
                              hipStream_t stream) {
    const float* x     = (const float*)d_in[0];
    const float* x0    = (const float*)d_in[1];
    const float* alpha = (const float*)d_in[2];
    const float* vals  = (const float*)d_in[3];
    const int*   rows  = (const int*)d_in[4];
    const int*   cols  = (const int*)d_in[5];

    const int n_nodes = in_sizes[2];
    const int n_edges = in_sizes[3];
    float* out = (float*)d_out;

    // 1) zero accumulator (out_size floats -> out_size/4 float4s)
    {
        const int n4 = out_size / 4;
        const int block = 256;
        int grid = (n4 + block - 1) / block;
        if (grid > 4096) grid = 4096; // grid-stride
        odefunc_zero<<<grid, block, 0, stream>>>((float4v*)out, n4);
    }

    // 2) edge scatter: 8 waves/block; ~8 edges per wave via grid-stride so
    //    the async double-buffered gather has work to overlap.
    {
        const int block = 256;                   // 8 wave32s
        const int waves_wanted = (n_edges + 7) / 8;
        int grid = (waves_wanted + 7) / 8;       // waves/8 -> blocks
        if (grid < 1) grid = 1;
        odefunc_spmm_scatter<<<grid, block, 0, stream>>>(
            vals, rows, cols, x, out, n_edges);
    }

    // 3) epilogue in place
    {
        const int total = n_nodes * CHUNKS_PER_NODE;
        const int block = 256;
        const int grid = (total + block - 1) / block;
        odefunc_finalize<<<grid, block, 0, stream>>>(x, x0, alpha, out, n_nodes);
    }
}